// TimingGNN_50757923504323
// MI455X (gfx1250) — compile-verified
//
#include <hip/hip_runtime.h>
#include <math.h>

#define NN 200000
#define EE 1000000
#define IN_DIM 16
#define HID 64
#define BN_EPS 1e-5f

typedef __attribute__((ext_vector_type(2))) float v2f;
typedef __attribute__((ext_vector_type(8))) float v8f;

// ---------------- degree / dinv ----------------
__global__ void k_init_deg(float* deg) {
    int i = blockIdx.x * blockDim.x + threadIdx.x;
    if (i < NN) deg[i] = 1.0f;
}

__global__ void k_add_deg(const long long* __restrict__ dst, float* __restrict__ deg) {
    int e = blockIdx.x * blockDim.x + threadIdx.x;
    if (e < EE) atomicAdd(&deg[dst[e]], 1.0f);
}

__global__ void k_dinv(float* deg) {
    int i = blockIdx.x * blockDim.x + threadIdx.x;
    if (i < NN) deg[i] = rsqrtf(deg[i]);
}

// ---------------- WMMA f32 GEMM: H[N,64] = A[N,K] @ W[K,64] ----------------
// Block = 128 threads (4 waves). Each wave owns one 16x16 output tile.
// blockIdx.x selects the 16-row tile; wave id selects the 16-col tile.
template <int K>
__global__ void k_gemm_wmma(const float* __restrict__ A, const float* __restrict__ W,
                            float* __restrict__ H) {
    const int wave  = threadIdx.x >> 5;
    const int lane  = threadIdx.x & 31;
    const int lrow  = lane & 15;          // M (and B column) within tile
    const int khalf = (lane >> 4) * 2;    // 0 for lanes 0-15, 2 for lanes 16-31
    const int row0  = blockIdx.x * 16;
    const int col0  = wave * 16;

    v8f acc = {};
    const float* Arow = A + (size_t)(row0 + lrow) * K;

#pragma unroll
    for (int k0 = 0; k0 < K; k0 += 4) {
        v2f a, b;
        // A 16x4 frag: VGPR0 = K=k0+khalf, VGPR1 = K=k0+khalf+1
        a.x = Arow[k0 + khalf];
        a.y = Arow[k0 + khalf + 1];
        // B 4x16 frag (W row-major [K,64]): same K striping, N = col0+lrow
        b.x = W[(size_t)(k0 + khalf) * HID + col0 + lrow];
        b.y = W[(size_t)(k0 + khalf + 1) * HID + col0 + lrow];
        acc = __builtin_amdgcn_wmma_f32_16x16x4_f32(
            /*neg_a=*/false, a, /*neg_b=*/false, b,
            /*c_mod=*/(short)0, acc, /*reuse_a=*/false, /*reuse_b=*/false);
    }

    // C/D layout: VGPR j -> row j (lanes 0-15) / row j+8 (lanes 16-31), col = N=lrow
    const int rbase = row0 + (lane >> 4) * 8;
    float* out = H + (size_t)rbase * HID + col0 + lrow;
#pragma unroll
    for (int j = 0; j < 8; ++j) out[(size_t)j * HID] = acc[j];
}

// ---------------- zero the aggregation buffer ----------------
__global__ void k_zero(float4* __restrict__ p, int n4) {
    int i = blockIdx.x * blockDim.x + threadIdx.x;
    if (i < n4) p[i] = make_float4(0.f, 0.f, 0.f, 0.f);
}

// ---------------- edge scatter: agg[dst] += h[src] * dinv[src]*dinv[dst] ----------------
// One wave per edge, each lane handles 2 consecutive features (64 total).
__global__ void k_aggregate(const float* __restrict__ H, const long long* __restrict__ src,
                            const long long* __restrict__ dst, const float* __restrict__ dinv,
                            float* __restrict__ Agg) {
    int t = blockIdx.x * blockDim.x + threadIdx.x;
    int e = t >> 5;
    if (e >= EE) return;
    int lane = t & 31;
    long long s = src[e], d = dst[e];
    float norm = dinv[s] * dinv[d];
    const float2 hv = *(const float2*)(H + (size_t)s * HID + lane * 2);
    float* ad = Agg + (size_t)d * HID + lane * 2;
    atomicAdd(&ad[0], hv.x * norm);
    atomicAdd(&ad[1], hv.y * norm);
}

// ---------------- fused self-loop + bias + BN + ReLU ----------------
// One wave per node; lane handles 2 features.
__global__ void k_post_bnrelu(const float* __restrict__ Agg, const float* __restrict__ H,
                              const float* __restrict__ dinv, const float* __restrict__ bias,
                              const float* __restrict__ g, const float* __restrict__ be,
                              const float* __restrict__ m, const float* __restrict__ v,
                              float* __restrict__ Out) {
    int t = blockIdx.x * blockDim.x + threadIdx.x;
    int n = t >> 5;
    if (n >= NN) return;
    int f = (t & 31) * 2;
    float di = dinv[n];
    float sl = di * di;
#pragma unroll
    for (int j = 0; j < 2; ++j) {
        int ff = f + j;
        size_t idx = (size_t)n * HID + ff;
        float xv = Agg[idx] + H[idx] * sl + bias[ff];
        float bnv = g[ff] * (xv - m[ff]) * rsqrtf(v[ff] + BN_EPS) + be[ff];
        Out[idx] = fmaxf(bnv, 0.0f);
    }
}

// ---------------- final: layer-3 post + residual + fc + sigmoid*10 ----------------
__global__ void k_final(const float* __restrict__ Agg, const float* __restrict__ H,
                        const float* __restrict__ dinv, const float* __restrict__ bias,
                        const float* __restrict__ g, const float* __restrict__ be,
                        const float* __restrict__ m, const float* __restrict__ v,
                        const float* __restrict__ resid, const float* __restrict__ fc_w,
                        const float* __restrict__ fc_b, float* __restrict__ out) {
    int t = blockIdx.x * blockDim.x + threadIdx.x;
    int n = t >> 5;
    if (n >= NN) return;
    int lane = t & 31;
    float di = dinv[n];
    float sl = di * di;
    float acc = 0.0f;
#pragma unroll
    for (int j = 0; j < 2; ++j) {
        int ff = lane * 2 + j;
        size_t idx = (size_t)n * HID + ff;
        float xv = Agg[idx] + H[idx] * sl + bias[ff];
        float bnv = g[ff] * (xv - m[ff]) * rsqrtf(v[ff] + BN_EPS) + be[ff];
        float hv = fmaxf(bnv, 0.0f) + resid[idx];
        acc += hv * fc_w[ff];
    }
    // wave32 reduction
#pragma unroll
    for (int off = 16; off > 0; off >>= 1) acc += __shfl_down(acc, off, 32);
    if (lane == 0) {
        float z = acc + fc_b[0];
        out[n] = 10.0f / (1.0f + expf(-z));
    }
}

extern "C" void kernel_launch(void* const* d_in, const int* in_sizes, int n_in,
                              void* d_out, int out_size, void* d_ws, size_t ws_size,
                              hipStream_t stream) {
    const float*     x    = (const float*)d_in[0];
    const long long* ei   = (const long long*)d_in[1];
    const long long* src  = ei;
    const long long* dst  = ei + EE;
    const float* W1 = (const float*)d_in[2];
    const float* b1 = (const float*)d_in[3];
    const float* W2 = (const float*)d_in[4];
    const float* b2 = (const float*)d_in[5];
    const float* W3 = (const float*)d_in[6];
    const float* b3 = (const float*)d_in[7];
    const float* fc_w = (const float*)d_in[8];
    const float* fc_b = (const float*)d_in[9];
    const float* g1 = (const float*)d_in[10]; const float* be1 = (const float*)d_in[11];
    const float* m1 = (const float*)d_in[12]; const float* v1  = (const float*)d_in[13];
    const float* g2 = (const float*)d_in[14]; const float* be2 = (const float*)d_in[15];
    const float* m2 = (const float*)d_in[16]; const float* v2  = (const float*)d_in[17];
    const float* g3 = (const float*)d_in[18]; const float* be3 = (const float*)d_in[19];
    const float* m3 = (const float*)d_in[20]; const float* v3  = (const float*)d_in[21];

    // workspace layout (floats)
    float* ws    = (float*)d_ws;
    float* dinv  = ws;                               // N
    float* hbuf  = dinv + NN;                        // N*64 (linear output)
    float* agg   = hbuf + (size_t)NN * HID;          // N*64 (aggregation)
    float* resid = agg  + (size_t)NN * HID;          // N*64 (layer-1 activation / residual)
    float* cur   = resid + (size_t)NN * HID;         // N*64 (layer-2 activation)

    const int rowTiles = NN / 16;        // 12500, exact
    const int n4       = NN * HID / 4;   // float4 count of an [N,64] buffer
    const int zeroBlk  = (n4 + 255) / 256;
    const int aggBlk   = (EE * 32 + 255) / 256;
    const int nodeBlk  = (NN * 32 + 255) / 256;

    // degree -> dinv
    k_init_deg<<<(NN + 255) / 256, 256, 0, stream>>>(dinv);
    k_add_deg<<<(EE + 255) / 256, 256, 0, stream>>>(dst, dinv);
    k_dinv<<<(NN + 255) / 256, 256, 0, stream>>>(dinv);

    // ---- layer 1: x[N,16] @ W1[16,64] ----
    k_gemm_wmma<IN_DIM><<<rowTiles, 128, 0, stream>>>(x, W1, hbuf);
    k_zero<<<zeroBlk, 256, 0, stream>>>((float4*)agg, n4);
    k_aggregate<<<aggBlk, 256, 0, stream>>>(hbuf, src, dst, dinv, agg);
    k_post_bnrelu<<<nodeBlk, 256, 0, stream>>>(agg, hbuf, dinv, b1, g1, be1, m1, v1, resid);

    // ---- layer 2: resid[N,64] @ W2[64,64] ----
    k_gemm_wmma<HID><<<rowTiles, 128, 0, stream>>>(resid, W2, hbuf);
    k_zero<<<zeroBlk, 256, 0, stream>>>((float4*)agg, n4);
    k_aggregate<<<aggBlk, 256, 0, stream>>>(hbuf, src, dst, dinv, agg);
    k_post_bnrelu<<<nodeBlk, 256, 0, stream>>>(agg, hbuf, dinv, b2, g2, be2, m2, v2, cur);

    // ---- layer 3: cur[N,64] @ W3[64,64], fused tail ----
    k_gemm_wmma<HID><<<rowTiles, 128, 0, stream>>>(cur, W3, hbuf);
    k_zero<<<zeroBlk, 256, 0, stream>>>((float4*)agg, n4);
    k_aggregate<<<aggBlk, 256, 0, stream>>>(hbuf, src, dst, dinv, agg);
    k_final<<<nodeBlk, 256, 0, stream>>>(agg, hbuf, dinv, b3, g3, be3, m3, v3,
                                         resid, fc_w, fc_b, (float*)d_out);
}